// MetaPathAdapter_89378269430326
// MI455X (gfx1250) — compile-verified
//
#include <hip/hip_runtime.h>
#include <hip/hip_bf16.h>

typedef __attribute__((ext_vector_type(2))) float v2f;
typedef __attribute__((ext_vector_type(8))) float v8f;

// ---------------------------------------------------------------------------
// Zero-fill (outputs are atomically accumulated; harness poisons d_out/d_ws)
// ---------------------------------------------------------------------------
__global__ void fill_zero_kernel(float* __restrict__ p, long long n) {
    long long i = (long long)blockIdx.x * blockDim.x + threadIdx.x;
    long long stride = (long long)gridDim.x * blockDim.x;
    for (; i < n; i += stride) p[i] = 0.0f;
}

// ---------------------------------------------------------------------------
// Dense projection Z[M,64] = X[M,K] @ W[K,64] via V_WMMA_F32_16X16X4_F32.
// fp32 WMMA keeps exact reference precision. One wave -> 16 rows x 64 cols.
//
// W is staged into LDS *pre-paired* along K:
//     sW[p*128 + 2*n + j] = W[(2p + j)*64 + n],  p = K-pair index, j in {0,1}
// so a B fragment (K=kb, kb+1 at column n) is ONE aligned ds_load_b64 that
// lands in an even-aligned VGPR pair -> no repacking movs before the WMMA.
//
// Fragment layouts (CDNA5 ISA 7.12.2, wave32):
//   A 16x4 f32 : lane l / l+16 -> M=l ; K = vgpr + 2*(lane>>4)
//   B 4x16 f32 : lane -> N = lane&15 ; K = vgpr + 2*(lane>>4)
//   C/D 16x16  : lane -> N = lane&15 ; M = vgpr + 8*(lane>>4)
// ---------------------------------------------------------------------------
template <int K>
__global__ __launch_bounds__(256) void gemm_wmma_f32_kernel(
    const float* __restrict__ X, const float* __restrict__ W,
    float* __restrict__ Z, int M) {
    __shared__ float sW[K * 64];   // K=256 -> 64KB (WGP has 320KB)

    for (int idx = threadIdx.x; idx < K * 64; idx += blockDim.x) {
        const int kk = idx >> 6;          // source row in W
        const int n  = idx & 63;          // source col in W
        sW[(kk >> 1) * 128 + 2 * n + (kk & 1)] = W[idx];
    }
    __syncthreads();

    const int wave = threadIdx.x >> 5;
    const int lane = threadIdx.x & 31;
    const int half = lane >> 4;   // 0: lanes 0-15, 1: lanes 16-31
    const int l    = lane & 15;

    const int row0 = (blockIdx.x * 8 + wave) * 16;
    if (row0 >= M) return;        // whole wave exits: EXEC stays all-ones for WMMA

    v8f acc[4] = {v8f{}, v8f{}, v8f{}, v8f{}};

    const float* xrow = X + (size_t)(row0 + l) * K + 2 * half;
    for (int k = 0; k < K; k += 4) {
        // A fragment: contiguous 8-byte global load per lane
        v2f av = *(const v2f*)(xrow + k);
        // pair index for this lane-half: kb = k + 2*half  ->  p = k/2 + half
        const float* bbase = sW + ((k >> 1) + half) * 128 + 2 * l;
#pragma unroll
        for (int nt = 0; nt < 4; ++nt) {
            v2f bv = *(const v2f*)(bbase + nt * 32);   // one ds_load_b64
            acc[nt] = __builtin_amdgcn_wmma_f32_16x16x4_f32(
                /*neg_a=*/false, av, /*neg_b=*/false, bv,
                /*c_mod=*/(short)0, acc[nt],
                /*reuse_a=*/false, /*reuse_b=*/false);
        }
    }

#pragma unroll
    for (int nt = 0; nt < 4; ++nt)
#pragma unroll
        for (int v = 0; v < 8; ++v)
            Z[(size_t)(row0 + v + 8 * half) * 64 + nt * 16 + l] = acc[nt][v];
}

// ---------------------------------------------------------------------------
// SpMM: out[row[e], :] += val[e] * zin[col[e], :]   (D = 64)
// 16 lanes per edge, each lane: one b128 gather + 4 native fp32 atomics.
// All operands fit in the 192MB L2, so the RMW traffic stays on-chip.
// Edge lists are single-use streams -> non-temporal loads keep the near
// caches free for the hot zin gather table.
// ---------------------------------------------------------------------------
__global__ __launch_bounds__(256) void spmm_scatter_kernel(
    const int* __restrict__ row, const int* __restrict__ col,
    const float* __restrict__ val, const float* __restrict__ zin,
    float* __restrict__ out, int E) {
    long long tid = (long long)blockIdx.x * blockDim.x + threadIdx.x;
    int e = (int)(tid >> 4);
    if (e >= E) return;
    int sub = (int)(tid & 15);

    int   c = __builtin_nontemporal_load(col + e);
    int   r = __builtin_nontemporal_load(row + e);
    float v = __builtin_nontemporal_load(val + e);

    float4 z4 = *((const float4*)(zin + (size_t)c * 64) + sub);
    float* dst = out + (size_t)r * 64 + sub * 4;
    unsafeAtomicAdd(dst + 0, v * z4.x);   // global_atomic_add_f32 (no return)
    unsafeAtomicAdd(dst + 1, v * z4.y);
    unsafeAtomicAdd(dst + 2, v * z4.z);
    unsafeAtomicAdd(dst + 3, v * z4.w);
}

// ---------------------------------------------------------------------------
// Orchestration. Sizes: N_P=200000, N_A=100000, D=64, F_P=256, F_A=128.
// d_out = [pap (N_P*64) | papap (N_P*64) | apa (N_A*64)]
// Workspace layout (floats):  zp[N_P*64] | za[N_A*64] | h[N_A*64]   (~103 MB)
//   h2 reuses h, g reuses zp (dead by then).
// ---------------------------------------------------------------------------
extern "C" void kernel_launch(void* const* d_in, const int* in_sizes, int n_in,
                              void* d_out, int out_size, void* d_ws, size_t ws_size,
                              hipStream_t stream) {
    const float* x_paper  = (const float*)d_in[0];
    const float* x_author = (const float*)d_in[1];
    const float* W_paper  = (const float*)d_in[2];
    const float* W_author = (const float*)d_in[3];
    const int*   pa_row   = (const int*)d_in[4];
    const int*   pa_col   = (const int*)d_in[5];
    const float* pa_val   = (const float*)d_in[6];
    const int*   ap_row   = (const int*)d_in[7];
    const int*   ap_col   = (const int*)d_in[8];
    const float* ap_val   = (const float*)d_in[9];

    const int N_P = in_sizes[0] / 256;   // 200000
    const int N_A = in_sizes[1] / 128;   // 100000
    const int E   = in_sizes[4];         // 1200000
    const int D   = 64;

    float* out   = (float*)d_out;
    float* pap   = out;
    float* papap = out + (size_t)N_P * D;
    float* apa   = out + (size_t)2 * N_P * D;

    float* zp = (float*)d_ws;                       // N_P*64
    float* za = zp + (size_t)N_P * D;               // N_A*64
    float* h  = za + (size_t)N_A * D;               // N_A*64 (also h2)
    float* g  = zp;                                 // reuse zp region (N_P*64)

    const long long out_elems = (long long)out_size;
    const long long hA = (long long)N_A * D;
    const long long hP = (long long)N_P * D;

    dim3 blk(256);
    const int fill_grid = 2048;
    const long long spmm_threads = 16LL * E;
    const int spmm_grid = (int)((spmm_threads + 255) / 256);

    // 0) zero outputs (accumulated via atomics)
    fill_zero_kernel<<<fill_grid, blk, 0, stream>>>(out, out_elems);

    // 1) projections (WMMA fp32)
    gemm_wmma_f32_kernel<256><<<(N_P / 16 + 7) / 8, blk, 0, stream>>>(x_paper, W_paper, zp, N_P);
    gemm_wmma_f32_kernel<128><<<(N_A / 16 + 7) / 8, blk, 0, stream>>>(x_author, W_author, za, N_A);

    // 2) metapath P-A-P:  h = pa @ zp ; pap = ap @ h
    fill_zero_kernel<<<fill_grid, blk, 0, stream>>>(h, hA);
    spmm_scatter_kernel<<<spmm_grid, blk, 0, stream>>>(pa_row, pa_col, pa_val, zp, h, E);
    spmm_scatter_kernel<<<spmm_grid, blk, 0, stream>>>(ap_row, ap_col, ap_val, h, pap, E);

    // 3) metapath P-A-P-A-P:  h2 = pa @ pap ; papap = ap @ h2   (h2 reuses h)
    fill_zero_kernel<<<fill_grid, blk, 0, stream>>>(h, hA);
    spmm_scatter_kernel<<<spmm_grid, blk, 0, stream>>>(pa_row, pa_col, pa_val, pap, h, E);
    spmm_scatter_kernel<<<spmm_grid, blk, 0, stream>>>(ap_row, ap_col, ap_val, h, papap, E);

    // 4) metapath A-P-A:  g = ap @ za ; apa = pa @ g   (g reuses zp slot)
    fill_zero_kernel<<<fill_grid, blk, 0, stream>>>(g, hP);
    spmm_scatter_kernel<<<spmm_grid, blk, 0, stream>>>(ap_row, ap_col, ap_val, za, g, E);
    spmm_scatter_kernel<<<spmm_grid, blk, 0, stream>>>(pa_row, pa_col, pa_val, g, apa, E);
}